// SGCMem_79577154060359
// MI455X (gfx1250) — compile-verified
//
#include <hip/hip_runtime.h>

#define IN_F  512
#define OUT_F 64

typedef __attribute__((ext_vector_type(2))) float v2f;
typedef __attribute__((ext_vector_type(8))) float v8f;

// ---------------- gcn_norm: degree + D^{-1/2} ----------------
__global__ void k_deg_init(float* __restrict__ deg, int n) {
    int i = blockIdx.x * blockDim.x + threadIdx.x;
    if (i < n) deg[i] = 1.0f;                      // self loop contributes 1
}

__global__ void k_deg_accum(const int* __restrict__ col, float* __restrict__ deg, int e) {
    int i = blockIdx.x * blockDim.x + threadIdx.x;
    if (i < e) atomicAdd(&deg[col[i]], 1.0f);
}

__global__ void k_dinv(float* __restrict__ deg, int n) {
    int i = blockIdx.x * blockDim.x + threadIdx.x;
    if (i < n) {
        float d = deg[i];
        deg[i] = (d > 0.0f) ? rsqrtf(d) : 0.0f;    // in-place: deg -> dinv
    }
}

// ---------------- h = x @ W^T + b via V_WMMA_F32_16X16X4_F32 ----------------
// Block: 128 threads = 4 waves. Block owns 16 rows; wave w owns cols [16w,16w+16).
// Wave accumulates D(16x16) over K=512 in steps of 4.
__global__ void __launch_bounds__(128)
k_gemm_wmma(const float* __restrict__ x, const float* __restrict__ W,
            const float* __restrict__ b, float* __restrict__ h)
{
    const int lane    = threadIdx.x & 31;
    const int wave    = threadIdx.x >> 5;          // 0..3 -> column tile
    const int l15     = lane & 15;
    const int halfSel = lane >> 4;                 // 0 -> K+{0,1}, 1 -> K+{2,3}
    const int rowTile = blockIdx.x * 16;
    const int colTile = wave * 16;

    // A: row = rowTile + l15 ; B: col = colTile + l15 (W is [OUT, IN] row-major)
    const float* __restrict__ xrow = x + (size_t)(rowTile + l15) * IN_F;
    const float* __restrict__ wrow = W + (size_t)(colTile + l15) * IN_F;

    v8f acc = {};
    #pragma unroll 4
    for (int k = 0; k < IN_F; k += 4) {
        const int kk = k + 2 * halfSel;
        v2f a  = *(const v2f*)(xrow + kk);          // global_load_b64
        v2f bb = *(const v2f*)(wrow + kk);
        acc = __builtin_amdgcn_wmma_f32_16x16x4_f32(
                  /*neg_a=*/false, a, /*neg_b=*/false, bb,
                  /*c_mod=*/(short)0, acc, /*reuse_a=*/false, /*reuse_b=*/false);
    }

    // D layout: VGPR r -> M = r + 8*halfSel, N = l15
    const int n     = colTile + l15;
    const int mBase = rowTile + 8 * halfSel;
    const float bias = b[n];
    #pragma unroll
    for (int r = 0; r < 8; ++r)
        h[(size_t)(mBase + r) * OUT_F + n] = acc[r] + bias;
}

// ---------------- SpMM hop ----------------
// dst[i,:] = dinv[i]^2 * src[i,:]   (self-loop term, also initializes dst)
__global__ void k_hop_init(const float* __restrict__ src, const float* __restrict__ dinv,
                           float* __restrict__ dst, int n64) {
    int i = blockIdx.x * blockDim.x + threadIdx.x;
    if (i < n64) {
        float d = dinv[i >> 6];
        dst[i] = d * d * src[i];
    }
}

// dst[col,:] += dinv[row]*dinv[col] * src[row,:]  — 32 threads/edge, 2 feats/thread
__global__ void k_hop_scatter(const int* __restrict__ row, const int* __restrict__ col,
                              const float* __restrict__ dinv,
                              const float* __restrict__ src, float* __restrict__ dst,
                              int e) {
    int t    = blockIdx.x * blockDim.x + threadIdx.x;
    int edge = t >> 5;
    int f2   = (t & 31) * 2;
    if (edge < e) {
        int r = row[edge], c = col[edge];
        float w = dinv[r] * dinv[c];
        const float* s = src + (size_t)r * OUT_F + f2;
        float*       d = dst + (size_t)c * OUT_F + f2;
        float2 sv = *(const float2*)s;              // global_load_b64, L2 hit (h fits in 192MB L2)
        atomicAdd(d + 0, w * sv.x);                 // global_atomic_add_f32 (no-return -> STOREcnt)
        atomicAdd(d + 1, w * sv.y);
    }
}

extern "C" void kernel_launch(void* const* d_in, const int* in_sizes, int n_in,
                              void* d_out, int out_size, void* d_ws, size_t ws_size,
                              hipStream_t stream) {
    const float* x  = (const float*)d_in[0];
    const float* W  = (const float*)d_in[1];
    const float* b  = (const float*)d_in[2];
    const int*   ei = (const int*)d_in[3];

    const int N = in_sizes[0] / IN_F;      // 100000
    const int E = in_sizes[3] / 2;         // 1600000
    const int* row = ei;                    // edge_index[0]
    const int* col = ei + E;                // edge_index[1]

    float* out  = (float*)d_out;

    // workspace: [dinv: N floats][align 256][bufA: N*64 floats]
    float* dinv = (float*)d_ws;
    size_t off  = ((size_t)N * sizeof(float) + 255) & ~(size_t)255;
    float* bufA = (float*)((char*)d_ws + off);

    const int TB = 256;

    // 1) degrees (incl. self loop) -> dinv
    k_deg_init <<<(N + TB - 1) / TB, TB, 0, stream>>>(dinv, N);
    k_deg_accum<<<(E + TB - 1) / TB, TB, 0, stream>>>(col, dinv, E);
    k_dinv     <<<(N + TB - 1) / TB, TB, 0, stream>>>(dinv, N);

    // 2) h0 = x @ W^T + b   (WMMA f32, one 16-row tile per block, exact grid)
    k_gemm_wmma<<<N / 16, 128, 0, stream>>>(x, W, b, out);

    // 3) 4 hops of normalized SpMM; ping-pong d_out <-> bufA (even #hops ends in d_out)
    const int n64    = N * OUT_F;
    const int initGB = (n64 + TB - 1) / TB;
    const int scatGB = (E * 32 + TB - 1) / TB;
    for (int hop = 0; hop < 4; ++hop) {
        const float* src = (hop & 1) ? bufA : out;
        float*       dst = (hop & 1) ? out  : bufA;
        k_hop_init   <<<initGB, TB, 0, stream>>>(src, dinv, dst, n64);
        k_hop_scatter<<<scatGB, TB, 0, stream>>>(row, col, dinv, src, dst, E);
    }
}